// TrajectoryLSTMModel_46961172415395
// MI455X (gfx1250) — compile-verified
//
#include <hip/hip_runtime.h>
#include <cmath>

// ---------------------------------------------------------------------------
// Problem sizes (fixed by the reference):
//   x: (16384, 20, 6) f32   -> LSTM scan runs over dim0 = 16384 SERIAL steps
//   per-step minibatch N=20, H=32, gates = 4H = 128
//   head: (16384,640)@(640,128) relu @(128,32) relu @(32,2)
// ---------------------------------------------------------------------------

typedef float v2f __attribute__((ext_vector_type(2)));
typedef float v8f __attribute__((ext_vector_type(8)));

__device__ __forceinline__ v8f wmma_f32_k4(v2f a, v2f b, v8f c) {
  // V_WMMA_F32_16X16X4_F32 : D(16x16,f32) = A(16x4,f32) * B(4x16,f32) + C
  return __builtin_amdgcn_wmma_f32_16x16x4_f32(
      false, a, false, b, (short)0, c, false, false);
}

__device__ __forceinline__ float sigf(float x) {
  return 1.0f / (1.0f + __expf(-x));
}

#define NSTEPS 16384
#define XROW   120      // 20*6 floats per step
#define HROW   640      // 20*32 floats per step

// ---------------------------------------------------------------------------
// Serial fused 2-layer LSTM chain. One workgroup, 256 threads = 8 waves.
// Each wave owns one 16-wide N-tile of the 128 gate columns and keeps its
// B-matrix WMMA fragments (layer1 K=40, layer2 K=64) resident in VGPRs for
// all 16384 steps. h-state flows through LDS; c-state lives in registers.
// ---------------------------------------------------------------------------
__global__ __launch_bounds__(256) void lstm_chain_kernel(
    const float* __restrict__ x,
    const float* __restrict__ Wih1, const float* __restrict__ Whh1,
    const float* __restrict__ bih1, const float* __restrict__ bhh1,
    const float* __restrict__ Wih2, const float* __restrict__ Whh2,
    const float* __restrict__ bih2, const float* __restrict__ bhh2,
    float* __restrict__ hbuf)   // (16384, 640) layer-2 hidden states
{
  __shared__ float A1[32][40];   // [x_t | h1_{t-1} | pad2], rows 20..31 zero
  __shared__ float A2[32][64];   // [h1_t | h2_{t-1}],       rows 20..31 zero
  __shared__ float G [32][128];  // gate pre-activations staging
  __shared__ float bs1[128];
  __shared__ float bs2[128];

  const int tid   = threadIdx.x;
  const int lane  = tid & 31;
  const int wid   = tid >> 5;
  const int nl    = lane & 15;            // N within tile (also M row for A)
  const int klane = (lane >> 4) << 1;     // K offset from lane half
  const int n0    = wid << 4;             // this wave's N-tile base
  const int rhalf = (lane >> 4) << 3;     // D-fragment row offset

  for (int i = tid; i < 32 * 40; i += 256) ((float*)A1)[i] = 0.0f;
  for (int i = tid; i < 32 * 64; i += 256) ((float*)A2)[i] = 0.0f;
  if (tid < 128) {
    bs1[tid] = bih1[tid] + bhh1[tid];
    bs2[tid] = bih2[tid] + bhh2[tid];
  }

  // ---- persistent, pre-swizzled B fragments (weights) -------------------
  // Layer1 Bcat (40 x 128): rows 0..5 = Wih1^T, rows 6..37 = Whh1^T, pad 0.
  v2f B1f[10];
#pragma unroll
  for (int kk = 0; kk < 10; ++kk) {
    const int n = n0 + nl;
    const int k = kk * 4 + klane;
    const int k1 = k + 1;
    B1f[kk].x = (k  < 6) ? Wih1[n * 6 + k ] : (k  < 38) ? Whh1[n * 32 + (k  - 6)] : 0.0f;
    B1f[kk].y = (k1 < 6) ? Wih1[n * 6 + k1] : (k1 < 38) ? Whh1[n * 32 + (k1 - 6)] : 0.0f;
  }
  // Layer2 Bcat (64 x 128): rows 0..31 = Wih2^T, rows 32..63 = Whh2^T.
  v2f B2f[16];
#pragma unroll
  for (int kk = 0; kk < 16; ++kk) {
    const int n = n0 + nl;
    const int k = kk * 4 + klane;
    const int k1 = k + 1;
    B2f[kk].x = (k  < 32) ? Wih2[n * 32 + k ] : Whh2[n * 32 + (k  - 32)];
    B2f[kk].y = (k1 < 32) ? Wih2[n * 32 + k1] : Whh2[n * 32 + (k1 - 32)];
  }

  // cell state lives in registers (never shared between threads)
  float c1r[3] = {0.0f, 0.0f, 0.0f};
  float c2r[3] = {0.0f, 0.0f, 0.0f};

  // x_t prefetched one step ahead into registers (threads 0..119)
  float xv = (tid < XROW) ? x[tid] : 0.0f;

#pragma unroll 1
  for (int t = 0; t < NSTEPS; ++t) {
    if (tid < XROW) A1[tid / 6][tid % 6] = xv;
    float xn = 0.0f;
    {
      const int tn = (t + 1 < NSTEPS) ? (t + 1) : t;
      if (tid < XROW) xn = x[(size_t)tn * XROW + tid];
      if (t + 2 < NSTEPS)
        __builtin_prefetch(x + (size_t)(t + 2) * XROW + lane, 0, 0);
    }
    __syncthreads();

    // ---------------- layer 1 gates: (32x40) @ (40x128) ------------------
    v8f acc0 = {0.f,0.f,0.f,0.f,0.f,0.f,0.f,0.f};
    v8f acc1 = {0.f,0.f,0.f,0.f,0.f,0.f,0.f,0.f};
#pragma unroll
    for (int kk = 0; kk < 10; ++kk) {
      const int k = kk * 4 + klane;
      v2f a0 = {A1[nl][k],      A1[nl][k + 1]};
      v2f a1 = {A1[16 + nl][k], A1[16 + nl][k + 1]};
      acc0 = wmma_f32_k4(a0, B1f[kk], acc0);
      acc1 = wmma_f32_k4(a1, B1f[kk], acc1);
    }
#pragma unroll
    for (int r = 0; r < 8; ++r) {
      G[rhalf + r][n0 + nl]      = acc0[r];
      G[16 + rhalf + r][n0 + nl] = acc1[r];
    }
    __syncthreads();

    // ---------------- layer 1 activations (640 elems) --------------------
#pragma unroll
    for (int j = 0; j < 3; ++j) {
      const int e = tid + 256 * j;
      if (e < HROW) {
        const int r = e >> 5, n = e & 31;
        const float gi = sigf (G[r][n]      + bs1[n]);
        const float gf = sigf (G[r][32 + n] + bs1[32 + n]);
        const float gg = tanhf(G[r][64 + n] + bs1[64 + n]);
        const float go = sigf (G[r][96 + n] + bs1[96 + n]);
        const float c  = gf * c1r[j] + gi * gg;
        c1r[j] = c;
        const float h = go * tanhf(c);
        A1[r][6 + n] = h;   // recurrent input for layer 1, next step
        A2[r][n]     = h;   // feed-forward input for layer 2, this step
      }
    }
    __syncthreads();

    // ---------------- layer 2 gates: (32x64) @ (64x128) ------------------
    acc0 = (v8f){0.f,0.f,0.f,0.f,0.f,0.f,0.f,0.f};
    acc1 = (v8f){0.f,0.f,0.f,0.f,0.f,0.f,0.f,0.f};
#pragma unroll
    for (int kk = 0; kk < 16; ++kk) {
      const int k = kk * 4 + klane;
      v2f a0 = {A2[nl][k],      A2[nl][k + 1]};
      v2f a1 = {A2[16 + nl][k], A2[16 + nl][k + 1]};
      acc0 = wmma_f32_k4(a0, B2f[kk], acc0);
      acc1 = wmma_f32_k4(a1, B2f[kk], acc1);
    }
#pragma unroll
    for (int r = 0; r < 8; ++r) {
      G[rhalf + r][n0 + nl]      = acc0[r];
      G[16 + rhalf + r][n0 + nl] = acc1[r];
    }
    __syncthreads();

    // ---------------- layer 2 activations + emit h2 ----------------------
#pragma unroll
    for (int j = 0; j < 3; ++j) {
      const int e = tid + 256 * j;
      if (e < HROW) {
        const int r = e >> 5, n = e & 31;
        const float gi = sigf (G[r][n]      + bs2[n]);
        const float gf = sigf (G[r][32 + n] + bs2[32 + n]);
        const float gg = tanhf(G[r][64 + n] + bs2[64 + n]);
        const float go = sigf (G[r][96 + n] + bs2[96 + n]);
        const float c  = gf * c2r[j] + gi * gg;
        c2r[j] = c;
        const float h = go * tanhf(c);
        A2[r][32 + n] = h;                       // recurrent input, next step
        hbuf[(size_t)t * HROW + e] = h;          // MLP input row t
      }
    }
    xv = xn;
    // loop-top __syncthreads orders these writes vs next iteration's reads
  }
}

// ---------------------------------------------------------------------------
// MLP head: (16384x640)@(640x128)+relu -> @(128x32)+relu -> @(32x2).
// 512 blocks x 32 rows; layer 1 on f32 WMMA, small layers on VALU.
// ---------------------------------------------------------------------------
__global__ __launch_bounds__(256) void mlp_head_kernel(
    const float* __restrict__ z,
    const float* __restrict__ W1, const float* __restrict__ b1,
    const float* __restrict__ W2, const float* __restrict__ b2,
    const float* __restrict__ W3, const float* __restrict__ b3,
    float* __restrict__ out)
{
  __shared__ float Z1[32][128];
  __shared__ float Z2[32][32];

  const int tid   = threadIdx.x;
  const int lane  = tid & 31;
  const int wid   = tid >> 5;
  const int nl    = lane & 15;
  const int klane = (lane >> 4) << 1;
  const int n0    = wid << 4;
  const int rhalf = (lane >> 4) << 3;
  const int row0  = blockIdx.x * 32;

  v8f acc0 = {0.f,0.f,0.f,0.f,0.f,0.f,0.f,0.f};
  v8f acc1 = {0.f,0.f,0.f,0.f,0.f,0.f,0.f,0.f};

  const float* zr0 = z  + (size_t)(row0 + nl)      * 640;
  const float* zr1 = z  + (size_t)(row0 + 16 + nl) * 640;
  const float* wr  = W1 + (size_t)(n0 + nl)        * 640;   // W1 row n = B col n

#pragma unroll 4
  for (int k0 = 0; k0 < 640; k0 += 4) {
    const int k = k0 + klane;
    const float2 bv  = *(const float2*)(wr  + k);
    const float2 a0v = *(const float2*)(zr0 + k);
    const float2 a1v = *(const float2*)(zr1 + k);
    v2f bf = {bv.x,  bv.y};
    v2f a0 = {a0v.x, a0v.y};
    v2f a1 = {a1v.x, a1v.y};
    acc0 = wmma_f32_k4(a0, bf, acc0);
    acc1 = wmma_f32_k4(a1, bf, acc1);
  }
  const float bb = b1[n0 + nl];
#pragma unroll
  for (int r = 0; r < 8; ++r) {
    Z1[rhalf + r][n0 + nl]      = fmaxf(acc0[r] + bb, 0.0f);
    Z1[16 + rhalf + r][n0 + nl] = fmaxf(acc1[r] + bb, 0.0f);
  }
  __syncthreads();

  // layer 2: 32x32 outputs, dot-128 each
#pragma unroll
  for (int oi = 0; oi < 4; ++oi) {
    const int e = tid + 256 * oi;       // 1024 outputs
    const int r = e >> 5, n = e & 31;
    float s = b2[n];
#pragma unroll 8
    for (int k = 0; k < 128; ++k) s += Z1[r][k] * W2[n * 128 + k];
    Z2[r][n] = fmaxf(s, 0.0f);
  }
  __syncthreads();

  // layer 3: 32x2 outputs, dot-32 each
  if (tid < 64) {
    const int r = tid >> 1, n = tid & 1;
    float s = b3[n];
#pragma unroll
    for (int k = 0; k < 32; ++k) s += Z2[r][k] * W3[n * 32 + k];
    out[(size_t)(row0 + r) * 2 + n] = s;
  }
}

// ---------------------------------------------------------------------------
extern "C" void kernel_launch(void* const* d_in, const int* in_sizes, int n_in,
                              void* d_out, int out_size, void* d_ws, size_t ws_size,
                              hipStream_t stream) {
  (void)in_sizes; (void)n_in; (void)out_size; (void)ws_size;

  const float* x    = (const float*)d_in[0];
  const float* Wih1 = (const float*)d_in[1];
  const float* Whh1 = (const float*)d_in[2];
  const float* bih1 = (const float*)d_in[3];
  const float* bhh1 = (const float*)d_in[4];
  const float* Wih2 = (const float*)d_in[5];
  const float* Whh2 = (const float*)d_in[6];
  const float* bih2 = (const float*)d_in[7];
  const float* bhh2 = (const float*)d_in[8];
  const float* W1   = (const float*)d_in[9];
  const float* b1   = (const float*)d_in[10];
  const float* W2   = (const float*)d_in[11];
  const float* b2   = (const float*)d_in[12];
  const float* W3   = (const float*)d_in[13];
  const float* b3   = (const float*)d_in[14];

  float* hbuf = (float*)d_ws;             // 16384*640 f32 = 40 MiB scratch
  float* out  = (float*)d_out;            // (16384, 2) f32

  lstm_chain_kernel<<<1, 256, 0, stream>>>(x, Wih1, Whh1, bih1, bhh1,
                                           Wih2, Whh2, bih2, bhh2, hbuf);
  mlp_head_kernel<<<16384 / 32, 256, 0, stream>>>(hbuf, W1, b1, W2, b2, W3, b3,
                                                  out);
}